// SCConv_27788438405860
// MI455X (gfx1250) — compile-verified
//
#include <hip/hip_runtime.h>
#include <hip/hip_bf16.h>

typedef __attribute__((ext_vector_type(2))) float v2f;
typedef __attribute__((ext_vector_type(8))) float v8f;

// ---------------------------------------------------------------------------
// Branch a, stage 1: conv3x3(pad1)+bias -> relu -> maxpool(2,2,s1)
// x1:(64,1,2,4900) -> Aact:(64,4899)
// ---------------------------------------------------------------------------
__global__ void k_branch_a(const float* __restrict__ x1, const float* __restrict__ w1,
                           const float* __restrict__ b1, float* __restrict__ Aact) {
  int j = blockIdx.x * blockDim.x + threadIdx.x;  // 0..4898
  int n = blockIdx.y;
  if (j >= 4899) return;
  const float* x = x1 + n * 2 * 4900;
  float w[9];
#pragma unroll
  for (int d = 0; d < 9; ++d) w[d] = w1[d];
  float bias = b1[0];
  float m = 0.f;  // pooled relu values are >= 0
#pragma unroll
  for (int i = 0; i < 2; ++i)
#pragma unroll
    for (int dj = 0; dj < 2; ++dj) {
      int jj = j + dj;  // conv col 0..4899
      float acc = bias;
#pragma unroll
      for (int di = 0; di < 3; ++di) {
        int r = i + di - 1;
        if (r < 0 || r > 1) continue;
#pragma unroll
        for (int dc = 0; dc < 3; ++dc) {
          int c = jj + dc - 1;
          if (c < 0 || c > 4899) continue;
          acc += x[r * 4900 + c] * w[di * 3 + dc];
        }
      }
      float rl = acc > 0.f ? acc : 0.f;
      m = rl > m ? rl : m;
    }
  Aact[n * 4899 + j] = m;
}

// ---------------------------------------------------------------------------
// Generic WMMA GEMM: C(64,N) = act( A(64,K) @ W(K,N) + bias ), fp32 WMMA 16x16x4
// block = 64 threads (2 waves); each wave: one 16-col tile x all 4 row tiles.
// grid.x = ceil(N/32)
// ---------------------------------------------------------------------------
__global__ void k_gemm_wmma(const float* __restrict__ A, const float* __restrict__ Wt,
                            const float* __restrict__ bias, float* __restrict__ C,
                            int N, int K, int do_relu) {
  int tid  = threadIdx.x;
  int wave = tid >> 5;
  int lane = tid & 31;
  int half = lane >> 4;   // which 16-lane half
  int l15  = lane & 15;
  int n    = blockIdx.x * 32 + wave * 16 + l15;
  bool nok = n < N;
  int ncl  = nok ? n : (N - 1);  // clamped address, value zeroed below

  v8f acc[4] = {};

  int Kmain = K & ~3;
  for (int k = 0; k < Kmain; k += 4) {
    int r0 = k + 2 * half;
    // weight stream: rows r0, r0+1 of W (HBM-bound path)
    v2f b;
    b.x = Wt[r0 * N + ncl];
    b.y = Wt[(r0 + 1) * N + ncl];
    if (!nok) { b.x = 0.f; b.y = 0.f; }
    // prefetch the weight stream 64 rows ahead (speculative; OOB dropped)
    __builtin_prefetch(&Wt[(r0 + 64) * N + ncl], 0, 1);
#pragma unroll
    for (int t = 0; t < 4; ++t) {
      int m = t * 16 + l15;
      v2f a;
      a.x = A[m * K + r0];
      a.y = A[m * K + r0 + 1];
      acc[t] = __builtin_amdgcn_wmma_f32_16x16x4_f32(false, a, false, b,
                                                     (short)0, acc[t], false, false);
    }
  }
  if (Kmain < K) {  // K tail (uniform branch; per-lane guards are selects)
    int r0 = Kmain + 2 * half;
    v2f b; b.x = 0.f; b.y = 0.f;
    if (nok) {
      if (r0 < K)     b.x = Wt[r0 * N + ncl];
      if (r0 + 1 < K) b.y = Wt[(r0 + 1) * N + ncl];
    }
#pragma unroll
    for (int t = 0; t < 4; ++t) {
      int m = t * 16 + l15;
      v2f a; a.x = 0.f; a.y = 0.f;
      if (r0 < K)     a.x = A[m * K + r0];
      if (r0 + 1 < K) a.y = A[m * K + r0 + 1];
      acc[t] = __builtin_amdgcn_wmma_f32_16x16x4_f32(false, a, false, b,
                                                     (short)0, acc[t], false, false);
    }
  }

  if (nok) {
    float bs = bias[n];
#pragma unroll
    for (int t = 0; t < 4; ++t)
#pragma unroll
      for (int v = 0; v < 8; ++v) {
        int m = t * 16 + v + 8 * half;
        float val = acc[t][v] + bs;
        if (do_relu) val = val > 0.f ? val : 0.f;
        C[m * N + n] = val;
      }
  }
}

// ---------------------------------------------------------------------------
// Branch a, stage 3: per-sample dynamic 3x3 conv (dynw = lw*a3), +lb, relu,
// maxpool(2,2,s2).  x1:(64,2,4900) -> cat[:, 0:2450]
// ---------------------------------------------------------------------------
__global__ void k_dynconv(const float* __restrict__ x1, const float* __restrict__ lw,
                          const float* __restrict__ lb, const float* __restrict__ A3,
                          float* __restrict__ cat) {
  int q = blockIdx.x * blockDim.x + threadIdx.x;  // 0..2449
  int n = blockIdx.y;
  if (q >= 2450) return;
  const float* x = x1 + n * 2 * 4900;
  float w[9];
#pragma unroll
  for (int d = 0; d < 9; ++d) w[d] = lw[n * 9 + d] * A3[n * 9 + d];
  float bias = lb[n];
  float m = 0.f;
#pragma unroll
  for (int i = 0; i < 2; ++i)
#pragma unroll
    for (int dj = 0; dj < 2; ++dj) {
      int jj = 2 * q + dj;
      float acc = bias;
#pragma unroll
      for (int di = 0; di < 3; ++di) {
        int r = i + di - 1;
        if (r < 0 || r > 1) continue;
#pragma unroll
        for (int dc = 0; dc < 3; ++dc) {
          int c = jj + dc - 1;
          if ((unsigned)c > 4899u) continue;
          acc += x[r * 4900 + c] * w[di * 3 + dc];
        }
      }
      float rl = acc > 0.f ? acc : 0.f;
      m = rl > m ? rl : m;
    }
  cat[n * 15506 + q] = m;
}

// ---------------------------------------------------------------------------
// Branch b, stage 1: conv2x2(pad1)+bias -> relu -> maxpool(2,2,s1)
// x2:(64,2,4900) -> ymid:(64,2,4900)
// ---------------------------------------------------------------------------
__global__ void k_b1(const float* __restrict__ x2, const float* __restrict__ w5,
                     const float* __restrict__ b5, float* __restrict__ ymid) {
  int j = blockIdx.x * blockDim.x + threadIdx.x;  // 0..4899
  int ni = blockIdx.y;                            // n*2 + i
  if (j >= 4900) return;
  int n = ni >> 1, i = ni & 1;
  const float* x = x2 + n * 2 * 4900;
  float bias = b5[0];
  float m = 0.f;
#pragma unroll
  for (int dr = 0; dr < 2; ++dr)
#pragma unroll
    for (int dc = 0; dc < 2; ++dc) {
      int r = i + dr;   // conv5 row 0..2
      int c = j + dc;   // conv5 col 0..4900
      float acc = bias;
#pragma unroll
      for (int u = 0; u < 2; ++u) {
        int rr = r + u - 1;
        if (rr < 0 || rr > 1) continue;
#pragma unroll
        for (int v = 0; v < 2; ++v) {
          int cc = c + v - 1;
          if ((unsigned)cc > 4899u) continue;
          acc += x[rr * 4900 + cc] * w5[u * 2 + v];
        }
      }
      float rl = acc > 0.f ? acc : 0.f;
      m = rl > m ? rl : m;
    }
  ymid[ni * 4900 + j] = m;
}

// ---------------------------------------------------------------------------
// Branch b, stage 2: conv(32,1,2,2,s2)+bias -> relu -> maxpool(1,6,s6)
// ymid:(64,2,4900) -> cat[:, 2450:15506]  (32*408 = 13056)
// ---------------------------------------------------------------------------
__global__ void k_b2(const float* __restrict__ ymid, const float* __restrict__ w6,
                     const float* __restrict__ b6, float* __restrict__ cat) {
  int idx = blockIdx.x * blockDim.x + threadIdx.x;  // k*408 + t
  int n = blockIdx.y;
  if (idx >= 32 * 408) return;
  int k = idx / 408, t = idx % 408;
  const float* y = ymid + n * 2 * 4900;
  float w00 = w6[k * 4 + 0], w01 = w6[k * 4 + 1];
  float w10 = w6[k * 4 + 2], w11 = w6[k * 4 + 3];
  float bias = b6[k];
  float m = 0.f;
#pragma unroll
  for (int s = 0; s < 6; ++s) {
    int c0 = 2 * (6 * t + s);  // <= 4894
    float acc = bias + y[c0] * w00 + y[c0 + 1] * w01
                     + y[4900 + c0] * w10 + y[4900 + c0 + 1] * w11;
    float rl = acc > 0.f ? acc : 0.f;
    m = rl > m ? rl : m;
  }
  cat[n * 15506 + 2450 + k * 408 + t] = m;
}

// ---------------------------------------------------------------------------
extern "C" void kernel_launch(void* const* d_in, const int* in_sizes, int n_in,
                              void* d_out, int out_size, void* d_ws, size_t ws_size,
                              hipStream_t stream) {
  const float* x1  = (const float*)d_in[0];
  const float* x2  = (const float*)d_in[1];
  const float* w1  = (const float*)d_in[2];
  const float* b1  = (const float*)d_in[3];
  const float* Wa  = (const float*)d_in[4];
  const float* ba  = (const float*)d_in[5];
  const float* Wb  = (const float*)d_in[6];
  const float* bb  = (const float*)d_in[7];
  const float* Wc  = (const float*)d_in[8];
  const float* bc  = (const float*)d_in[9];
  const float* lw  = (const float*)d_in[10];
  const float* lb  = (const float*)d_in[11];
  const float* w5  = (const float*)d_in[12];
  const float* b5  = (const float*)d_in[13];
  const float* w6  = (const float*)d_in[14];
  const float* b6  = (const float*)d_in[15];
  const float* Wf1 = (const float*)d_in[16];
  const float* bf1 = (const float*)d_in[17];
  const float* Wf2 = (const float*)d_in[18];
  const float* bf2 = (const float*)d_in[19];
  const float* Wf3 = (const float*)d_in[20];
  const float* bf3 = (const float*)d_in[21];

  float* ws   = (float*)d_ws;
  float* Aact = ws;                 // 64*4899   = 313536
  float* H1   = Aact + 313536;      // 64*1000   =  64000
  float* H2   = H1 + 64000;         // 64*500    =  32000
  float* A3   = H2 + 32000;         // 64*9      =    576
  float* ymid = A3 + 576;           // 64*2*4900 = 627200
  float* cat  = ymid + 627200;      // 64*15506  = 992384
  float* Z1   = cat + 992384;       // 64*8000   = 512000
  float* Z2   = Z1 + 512000;        // 64*2000   = 128000  (total ~10.7 MB)

  // branch a
  k_branch_a<<<dim3((4899 + 255) / 256, 64), 256, 0, stream>>>(x1, w1, b1, Aact);
  k_gemm_wmma<<<dim3((1000 + 31) / 32), 64, 0, stream>>>(Aact, Wa, ba, H1, 1000, 4899, 1);
  k_gemm_wmma<<<dim3((500 + 31) / 32), 64, 0, stream>>>(H1, Wb, bb, H2, 500, 1000, 1);
  k_gemm_wmma<<<1, 64, 0, stream>>>(H2, Wc, bc, A3, 9, 500, 0);
  k_dynconv<<<dim3((2450 + 255) / 256, 64), 256, 0, stream>>>(x1, lw, lb, A3, cat);
  // branch b
  k_b1<<<dim3((4900 + 255) / 256, 128), 256, 0, stream>>>(x2, w5, b5, ymid);
  k_b2<<<dim3((32 * 408 + 255) / 256, 64), 256, 0, stream>>>(ymid, w6, b6, cat);
  // fused head (HBM-bound weight stream)
  k_gemm_wmma<<<dim3((8000 + 31) / 32), 64, 0, stream>>>(cat, Wf1, bf1, Z1, 8000, 15506, 1);
  k_gemm_wmma<<<dim3((2000 + 31) / 32), 64, 0, stream>>>(Z1, Wf2, bf2, Z2, 2000, 8000, 1);
  k_gemm_wmma<<<1, 64, 0, stream>>>(Z2, Wf3, bf3, (float*)d_out, 2, 2000, 0);
}